// qcircuit_62079457296674
// MI455X (gfx1250) — compile-verified
//
#include <hip/hip_runtime.h>

typedef float v2f __attribute__((ext_vector_type(2)));
typedef float v8f __attribute__((ext_vector_type(8)));

// ---------------- complex helpers ----------------
__device__ __forceinline__ float2 cmul(float2 a, float2 b) {
    return make_float2(a.x * b.x - a.y * b.y, a.x * b.y + a.y * b.x);
}
__device__ __forceinline__ float2 cadd(float2 a, float2 b) {
    return make_float2(a.x + b.x, a.y + b.y);
}

// ring_of_cnot(4) acting on a 4-bit basis index.
// multi_kron puts qubit q at bit (3-q). CNOT(i,(i+1)%4) applied for i=0..3 in order.
__device__ __forceinline__ int ring_perm(int x) {
#pragma unroll
    for (int i = 0; i < 4; ++i) {
        int cbit = 3 - i;
        int tbit = 3 - ((i + 1) & 3);
        if ((x >> cbit) & 1) x ^= (1 << tbit);
    }
    return x;
}

// ---------------------------------------------------------------------------
// Kernel 1: build Tr = Re( U^H (X⊗X⊗X⊗X) U ), a real symmetric 16x16 matrix.
// One block of 256 threads, one thread per (r,c) matrix element.
// ---------------------------------------------------------------------------
__global__ __launch_bounds__(256) void build_T_kernel(const float* __restrict__ w,
                                                      float* __restrict__ Tr) {
    __shared__ float2 u[3][4][2][2];   // per layer, per qubit: Ry@Rz@Ry
    __shared__ float2 Y[3][16][16];    // yzy layers (kron of the four u's)
    __shared__ float2 Abuf[16][16];
    __shared__ float2 Bbuf[16][16];

    const int tid = threadIdx.x;
    const int r = tid >> 4, c = tid & 15;

    if (tid == 0) {
        for (int l = 0; l < 3; ++l)
            for (int q = 0; q < 4; ++q) {
                float t0 = w[12 * l + 3 * q + 0];
                float t1 = w[12 * l + 3 * q + 1];
                float t2 = w[12 * l + 3 * q + 2];
                float c0 = cosf(0.5f * t0), s0 = sinf(0.5f * t0);
                float cz = cosf(0.5f * t1), sz = sinf(0.5f * t1);
                float c2 = cosf(0.5f * t2), s2 = sinf(0.5f * t2);
                // m = Rz(t1) @ Ry(t0):
                //   row0 = e^{-i t1/2} * ( c0, -s0 ),  row1 = e^{+i t1/2} * ( s0, c0 )
                float2 m00 = make_float2( cz * c0, -sz * c0);
                float2 m01 = make_float2(-cz * s0,  sz * s0);
                float2 m10 = make_float2( cz * s0,  sz * s0);
                float2 m11 = make_float2( cz * c0,  sz * c0);
                // u = Ry(t2) @ m,  Ry rows: (c2,-s2) , (s2,c2)  (real)
                u[l][q][0][0] = make_float2(c2 * m00.x - s2 * m10.x, c2 * m00.y - s2 * m10.y);
                u[l][q][0][1] = make_float2(c2 * m01.x - s2 * m11.x, c2 * m01.y - s2 * m11.y);
                u[l][q][1][0] = make_float2(s2 * m00.x + c2 * m10.x, s2 * m00.y + c2 * m10.y);
                u[l][q][1][1] = make_float2(s2 * m01.x + c2 * m11.x, s2 * m01.y + c2 * m11.y);
            }
    }
    __syncthreads();

    // Y_l[r][c] = prod_q u[l][q][bit(r,3-q)][bit(c,3-q)]
#pragma unroll
    for (int l = 0; l < 3; ++l) {
        float2 v = u[l][0][(r >> 3) & 1][(c >> 3) & 1];
        v = cmul(v, u[l][1][(r >> 2) & 1][(c >> 2) & 1]);
        v = cmul(v, u[l][2][(r >> 1) & 1][(c >> 1) & 1]);
        v = cmul(v, u[l][3][r & 1][c & 1]);
        Y[l][r][c] = v;
    }
    __syncthreads();

    // U = Y2 @ R @ Y1 @ R @ Y0
    Abuf[r][c] = Y[0][r][c];
    __syncthreads();
    Bbuf[ring_perm(r)][c] = Abuf[r][c];     // B = R @ A (row scatter, bijective)
    __syncthreads();
    {
        float2 s = make_float2(0.f, 0.f);
        for (int k = 0; k < 16; ++k) s = cadd(s, cmul(Y[1][r][k], Bbuf[k][c]));
        Abuf[r][c] = s;
    }
    __syncthreads();
    Bbuf[ring_perm(r)][c] = Abuf[r][c];
    __syncthreads();
    {
        float2 s = make_float2(0.f, 0.f);
        for (int k = 0; k < 16; ++k) s = cadd(s, cmul(Y[2][r][k], Bbuf[k][c]));
        Abuf[r][c] = s;
    }
    __syncthreads();

    // T[r][c] = sum_k conj(U[k][r]) * U[15^k][c];  keep real part only.
    float s = 0.f;
    for (int k = 0; k < 16; ++k) {
        float2 a = Abuf[k][r];
        float2 b = Abuf[15 ^ k][c];
        s += a.x * b.x + a.y * b.y;
    }
    Tr[r * 16 + c] = s;
}

// ---------------------------------------------------------------------------
// One 16-column tile: Y = Tr @ S via 4 chained V_WMMA_F32_16X16X4_F32,
// then acc += sum(S .* Y) over this lane's 8 D elements.
// ---------------------------------------------------------------------------
__device__ __forceinline__ void tile_qf(const float* __restrict__ row,
                                        int col, int rmask, int sh,
                                        int kb, int mof,
                                        v2f a0, v2f a1, v2f a2, v2f a3,
                                        float& acc) {
    const int base = ((col & ~rmask) << 4) | (col & rmask);
    v8f d = {};
    v2f b;
    b.x = row[base + (( 0 + kb) << sh)];
    b.y = row[base + (( 1 + kb) << sh)];
    d = __builtin_amdgcn_wmma_f32_16x16x4_f32(false, a0, false, b, (short)0, d, false, false);
    b.x = row[base + (( 4 + kb) << sh)];
    b.y = row[base + (( 5 + kb) << sh)];
    d = __builtin_amdgcn_wmma_f32_16x16x4_f32(false, a1, false, b, (short)0, d, false, false);
    b.x = row[base + (( 8 + kb) << sh)];
    b.y = row[base + (( 9 + kb) << sh)];
    d = __builtin_amdgcn_wmma_f32_16x16x4_f32(false, a2, false, b, (short)0, d, false, false);
    b.x = row[base + ((12 + kb) << sh)];
    b.y = row[base + ((13 + kb) << sh)];
    d = __builtin_amdgcn_wmma_f32_16x16x4_f32(false, a3, false, b, (short)0, d, false, false);
#pragma unroll
    for (int rr = 0; rr < 8; ++rr)
        acc += row[base + ((rr + mof) << sh)] * d[rr];
}

// ---------------------------------------------------------------------------
// Kernel 2: one workgroup per batch row. Stage the full 256 KB row in LDS,
// then compute all 13 quadratic forms with V_WMMA_F32_16X16X4_F32.
// ---------------------------------------------------------------------------
__global__ __launch_bounds__(256) void shadow_expect_kernel(const float* __restrict__ vb,
                                                            const float* __restrict__ Tr,
                                                            float* __restrict__ out) {
    extern __shared__ float row[];      // 65536 floats = 256 KB dynamic LDS
    __shared__ float accs[13];
    __shared__ float norm2s;

    const int tid  = threadIdx.x;
    const int lane = tid & 31;
    const int wave = tid >> 5;          // 8 waves of 32

    if (tid < 13) accs[tid] = 0.0f;
    if (tid == 32) norm2s = 0.0f;
    __syncthreads();

    // ---- stage row into LDS, accumulate ||v||^2 on the way ----
    const float4* src = (const float4*)(vb + (size_t)blockIdx.x * 65536);
    float4* dst = (float4*)row;
    float nrm = 0.0f;
#pragma unroll 4
    for (int i = tid; i < 16384; i += 256) {
        float4 x = src[i];
        dst[i] = x;
        nrm += x.x * x.x + x.y * x.y + x.z * x.z + x.w * x.w;
    }
#pragma unroll
    for (int off = 16; off > 0; off >>= 1) nrm += __shfl_xor(nrm, off, 32);
    if (lane == 0) atomicAdd(&norm2s, nrm);

    // ---- A = Tr, resident in registers (ISA 16x4 f32 A layout, 4 K-chunks) ----
    const int M    = lane & 15;
    const int koff = (lane & 16) ? 2 : 0;   // K offset within a chunk for hi half-wave
    v2f a0 = { Tr[M * 16 + koff +  0], Tr[M * 16 + koff +  1] };
    v2f a1 = { Tr[M * 16 + koff +  4], Tr[M * 16 + koff +  5] };
    v2f a2 = { Tr[M * 16 + koff +  8], Tr[M * 16 + koff +  9] };
    v2f a3 = { Tr[M * 16 + koff + 12], Tr[M * 16 + koff + 13] };

    __syncthreads();   // row fully in LDS, norm2s complete

    const int colu = lane & 15;             // N index of this lane (B and D layouts)
    const int kb   = koff;                  // B-matrix K offset (same pattern as A)
    const int mof  = (lane & 16) ? 8 : 0;   // D-matrix M offset for hi half-wave

    for (int p = 0; p < 13; ++p) {
        const int sh    = 12 - p;           // right = 2^sh; 4-bit field at bits [sh, sh+4)
        const int rmask = (1 << sh) - 1;
        float acc = 0.0f;
        // 4096 columns / 16 per tile = 256 tiles; 32 per wave; 2 independent
        // tiles per iteration -> two independent WMMA accumulate chains.
        for (int t = 0; t < 32; t += 2) {
            const int col0 = (wave + t * 8) * 16 + colu;
            const int col1 = col0 + 8 * 16;
            tile_qf(row, col0, rmask, sh, kb, mof, a0, a1, a2, a3, acc);
            tile_qf(row, col1, rmask, sh, kb, mof, a0, a1, a2, a3, acc);
        }
#pragma unroll
        for (int off = 16; off > 0; off >>= 1) acc += __shfl_xor(acc, off, 32);
        if (lane == 0) atomicAdd(&accs[p], acc);
    }
    __syncthreads();
    if (tid < 13) out[blockIdx.x * 13 + tid] = accs[tid] / norm2s;
}

// ---------------------------------------------------------------------------
extern "C" void kernel_launch(void* const* d_in, const int* in_sizes, int n_in,
                              void* d_out, int out_size, void* d_ws, size_t ws_size,
                              hipStream_t stream) {
    (void)in_sizes; (void)n_in; (void)out_size; (void)ws_size;
    const float* vb = (const float*)d_in[0];   // (512, 65536) fp32
    const float* w  = (const float*)d_in[1];   // (36,) fp32
    float* Tr  = (float*)d_ws;                 // 256 floats scratch
    float* out = (float*)d_out;                // (512, 13) fp32

    hipFuncSetAttribute((const void*)shadow_expect_kernel,
                        hipFuncAttributeMaxDynamicSharedMemorySize,
                        65536 * (int)sizeof(float));

    build_T_kernel<<<1, 256, 0, stream>>>(w, Tr);
    shadow_expect_kernel<<<512, 256, 65536 * sizeof(float), stream>>>(vb, Tr, out);
}